// Embedder_86423331930547
// MI455X (gfx1250) — compile-verified
//
#include <hip/hip_runtime.h>
#include <stdint.h>
#include <math.h>

#define NB     8192      // B rows
#define KNZ    128       // nonzeros per row
#define GN     20000     // genes
#define HD     256       // hidden dim
#define CHUNK  16        // gathered rows per pipeline stage (= TDM gather limit)
#define NCHUNK (KNZ / CHUNK)
#define LN_EPS 1e-5f

typedef unsigned int v4u __attribute__((ext_vector_type(4)));
typedef int          v8i __attribute__((ext_vector_type(8)));
typedef int          v4i __attribute__((ext_vector_type(4)));

// ---------------------------------------------------------------------------
// Issue one TDM gather-mode descriptor: 16 rows of emb (256 f32 each, selected
// by 16-bit row indices) -> contiguous 16KB tile in LDS.  D# layout per
// CDNA5 ISA.
// ---------------------------------------------------------------------------
__device__ __forceinline__ void tdm_gather16(unsigned long long gbase,
                                             unsigned lds_addr,
                                             const int* __restrict__ scol) {
    v4u g0;
    g0.x = 1u | (1u << 31);                                  // count=1, gather_mode=1, 16-bit idx
    g0.y = (unsigned)__builtin_amdgcn_readfirstlane((int)lds_addr);
    g0.z = (unsigned)(gbase & 0xFFFFFFFFull);                // global_addr[31:0]
    g0.w = (unsigned)((gbase >> 32) & 0x1FFFFFFull) | (2u << 30); // addr[56:32] | type=2

    v8i g1;
    g1[0] = 0x20000;             // workgroup_mask=0, data_size=2 (4B)
    g1[1] = (int)(256u << 16);   // tensor_dim0 = 256  (bits 63:48)
    g1[2] = (int)(20000u << 16); // tensor_dim1 = G    (bits 95:80)
    g1[3] = (int)(256u << 16);   // tile_dim0  = 256   (bits 127:112)
    g1[4] = CHUNK;               // tile_dim1  = #valid indices = 16
    g1[5] = 256;                 // tensor_dim0_stride = 256 (bits 191:160)
    g1[6] = 0;
    g1[7] = 0;

    v4i g2, g3;
    #pragma unroll
    for (int j = 0; j < 4; ++j) {
        int lo = scol[2 * j], hi = scol[2 * j + 1];
        g2[j] = __builtin_amdgcn_readfirstlane((lo & 0xFFFF) | (hi << 16));
    }
    #pragma unroll
    for (int j = 0; j < 4; ++j) {
        int lo = scol[8 + 2 * j], hi = scol[8 + 2 * j + 1];
        g3[j] = __builtin_amdgcn_readfirstlane((lo & 0xFFFF) | (hi << 16));
    }

    const v8i g4 = (v8i)(0);     // extra arg in clang-23 6-arg form: zero-filled

    __builtin_amdgcn_tensor_load_to_lds(g0, g1, g2, g3, g4, 0);
}

// ---------------------------------------------------------------------------
// Fused kernel: TDM-gathered GEMV + exact GELU + LayerNorm.
// One block per row b; 256 threads (8 wave32); thread t owns h = t.
// ---------------------------------------------------------------------------
__global__ __launch_bounds__(HD) void embedder_fused_kernel(
    const float* __restrict__ x_vals,   // [B,K]
    const int*   __restrict__ x_cols,   // [B,K]
    const float* __restrict__ emb,      // [G,H]
    const float* __restrict__ ln_gamma, // [H]
    const float* __restrict__ ln_beta,  // [H]
    float*       __restrict__ out)      // [B,H]
{
    __shared__ __align__(16) float fbuf[2][CHUNK * HD];  // 2 x 16KB gather tiles
    __shared__ __align__(16) float s_vals[KNZ];
    __shared__ __align__(16) int   s_cols[KNZ];
    __shared__ float s_red[16];

    const int t = threadIdx.x;
    const int b = blockIdx.x;

    // Stage this row's nonzero values / column indices via async DMA to LDS.
    if (t < 32) {            // 32 x B128 = 512B of x_vals
        const unsigned long long ga =
            (unsigned long long)(const void*)x_vals + (unsigned long long)b * 512ull
            + (unsigned long long)t * 16ull;
        const unsigned la = (unsigned)(uintptr_t)(&s_vals[0]) + (unsigned)t * 16u;
        asm volatile("global_load_async_to_lds_b128 %0, %1, off"
                     :: "v"(la), "v"(ga) : "memory");
    } else if (t < 64) {     // 32 x B128 = 512B of x_cols
        const unsigned long long ga =
            (unsigned long long)(const void*)x_cols + (unsigned long long)b * 512ull
            + (unsigned long long)(t - 32) * 16ull;
        const unsigned la = (unsigned)(uintptr_t)(&s_cols[0]) + (unsigned)(t - 32) * 16u;
        asm volatile("global_load_async_to_lds_b128 %0, %1, off"
                     :: "v"(la), "v"(ga) : "memory");
    }
    asm volatile("s_wait_asynccnt 0x0" ::: "memory");
    __syncthreads();

    const unsigned long long emb_base = (unsigned long long)(const void*)emb;
    const unsigned lds_buf0 = (unsigned)(uintptr_t)(&fbuf[0][0]);
    const unsigned lds_buf1 = (unsigned)(uintptr_t)(&fbuf[1][0]);

    // Prologue: wave 0 issues the chunk-0 gather and waits for it.
    if (t < 32) {
        tdm_gather16(emb_base, lds_buf0, &s_cols[0]);
        __builtin_amdgcn_s_wait_tensorcnt(0);
    }
    __syncthreads();

    float acc = 0.0f;

    for (int c = 0; c < NCHUNK; ++c) {
        // Wave 0 prefetches chunk c+1 into the other buffer (overlaps compute).
        if (c + 1 < NCHUNK && t < 32) {
            tdm_gather16(emb_base, ((c + 1) & 1) ? lds_buf1 : lds_buf0,
                         &s_cols[(c + 1) * CHUNK]);
        }

        const float* fb = &fbuf[c & 1][0];
        #pragma unroll
        for (int r = 0; r < CHUNK; ++r) {
            acc = fmaf(s_vals[c * CHUNK + r], fb[r * HD + t], acc);
        }

        if (c + 1 < NCHUNK) {
            if (t < 32) __builtin_amdgcn_s_wait_tensorcnt(0);
            __syncthreads();  // chunk c+1 tile visible to all waves; everyone
                              // is done reading the buffer reused at c+2
        }
    }

    // Exact (erf) GELU, matching approximate=False.
    const float g = 0.5f * acc * (1.0f + erff(acc * 0.70710678118654752440f));

    // LayerNorm over the 256 lanes: wave32 shuffle reduce, then LDS combine.
    float s1 = g, s2 = g * g;
    #pragma unroll
    for (int o = 16; o > 0; o >>= 1) {
        s1 += __shfl_down(s1, o, 32);
        s2 += __shfl_down(s2, o, 32);
    }
    const int wave = t >> 5, lane = t & 31;
    if (lane == 0) { s_red[wave] = s1; s_red[8 + wave] = s2; }
    __syncthreads();
    if (t == 0) {
        float a1 = 0.0f, a2 = 0.0f;
        #pragma unroll
        for (int w = 0; w < 8; ++w) { a1 += s_red[w]; a2 += s_red[8 + w]; }
        const float mu  = a1 * (1.0f / HD);
        const float var = a2 * (1.0f / HD) - mu * mu;
        s_red[0] = mu;
        s_red[1] = rsqrtf(var + LN_EPS);
    }
    __syncthreads();
    const float mu = s_red[0], rstd = s_red[1];

    out[(size_t)b * HD + t] = (g - mu) * rstd * ln_gamma[t] + ln_beta[t];
}

// ---------------------------------------------------------------------------
// gene_idx passthrough (second tuple output), appended after the [B,H] floats.
// ---------------------------------------------------------------------------
__global__ void copy_gidx_i64_kernel(const long long* __restrict__ g,
                                     long long* __restrict__ dst, int n) {
    int i = blockIdx.x * blockDim.x + threadIdx.x;
    if (i < n) dst[i] = g[i];
}

__global__ void copy_gidx_i32_kernel(const long long* __restrict__ g,
                                     int* __restrict__ dst, int n) {
    int i = blockIdx.x * blockDim.x + threadIdx.x;
    if (i < n) dst[i] = (int)g[i];
}

// ---------------------------------------------------------------------------
extern "C" void kernel_launch(void* const* d_in, const int* in_sizes, int n_in,
                              void* d_out, int out_size, void* d_ws, size_t ws_size,
                              hipStream_t stream) {
    (void)in_sizes; (void)n_in; (void)d_ws; (void)ws_size;

    const float*     x_vals   = (const float*)d_in[0];
    const int*       x_cols   = (const int*)d_in[1];
    const long long* gene_idx = (const long long*)d_in[2];
    const float*     emb      = (const float*)d_in[3];
    const float*     ln_gamma = (const float*)d_in[4];
    const float*     ln_beta  = (const float*)d_in[5];
    float*           out      = (float*)d_out;

    embedder_fused_kernel<<<NB, HD, 0, stream>>>(x_vals, x_cols, emb,
                                                 ln_gamma, ln_beta, out);

    // Append gene_idx after the B*H floats; slot width inferred from out_size.
    char* tail_bytes = (char*)d_out + (size_t)NB * HD * sizeof(float);
    const int tail_elems = out_size - NB * HD;
    const int blocks = (GN + 255) / 256;
    if (tail_elems >= 2 * GN) {
        copy_gidx_i64_kernel<<<blocks, 256, 0, stream>>>(gene_idx,
                                                         (long long*)tail_bytes, GN);
    } else if (tail_elems >= GN) {
        copy_gidx_i32_kernel<<<blocks, 256, 0, stream>>>(gene_idx,
                                                         (int*)tail_bytes, GN);
    }
}